// PaiNNMessageBlock_2619930050847
// MI455X (gfx1250) — compile-verified
//
#include <hip/hip_runtime.h>

// PaiNN message block for MI455X (gfx1250, wave32).
// phi GEMMs via V_WMMA_F32_16X16X4_F32 (fp32-exact), edge stage via VALU +
// hardware fp32 global atomics. S-tile staging uses GLOBAL_LOAD_ASYNC_TO_LDS
// (ASYNCcnt-tracked) — builtin confirmed present, param type v4i* per AS.

#define NF   128      // F
#define N3F  384      // 3F
#define NR   20       // R

typedef float v2f __attribute__((ext_vector_type(2)));
typedef float v8f __attribute__((ext_vector_type(8)));
typedef int   v4i __attribute__((ext_vector_type(4)));

#define TILE_M     16
#define LDS_STRIDE 132          // 128 + 4 pad: lane m*132 -> bank 4m, conflict-free
#define WPB        4            // waves (tiles) per block

#if __has_builtin(__builtin_amdgcn_global_load_async_to_lds_b128) && \
    __has_builtin(__builtin_amdgcn_s_wait_asynccnt)
#define USE_ASYNC_LDS 1
typedef __attribute__((address_space(1))) v4i* g4i_t;   // global int4*
typedef __attribute__((address_space(3))) v4i* l4i_t;   // LDS int4*
#else
#define USE_ASYNC_LDS 0
#endif

// fast SiLU: x * rcp(1+exp(-x)); v_rcp_f32 (~1 ulp) instead of the
// correctly-rounded v_div_scale/newton chain the plain '/' emits.
__device__ __forceinline__ float fast_silu(float x) {
    return x * __builtin_amdgcn_rcpf(1.0f + __expf(-x));
}

// ---------------------------------------------------------------------------
// Kernel 1: phi[n, 384] = silu(S[n,128] @ W1 + b1) @ W2 + b2
// One wave handles one 16-row node tile. All matrices fp32, WMMA 16x16x4.
// ---------------------------------------------------------------------------
__global__ __launch_bounds__(32 * WPB) void painn_phi_wmma(
    const float* __restrict__ S,  const float* __restrict__ W1,
    const float* __restrict__ b1, const float* __restrict__ W2,
    const float* __restrict__ b2, float* __restrict__ phi, int n_nodes)
{
    __shared__ float lds[WPB * 2 * TILE_M * LDS_STRIDE];

    const int wave = threadIdx.x >> 5;
    const int lane = threadIdx.x & 31;
    const int half = lane >> 4;        // selects K pair {0,1} vs {2,3}
    const int l15  = lane & 15;        // A-row m / B,C column n (within tile)

    const int n_tiles = n_nodes / TILE_M;          // 625 (exact)
    int tile = blockIdx.x * WPB + wave;
    if (tile >= n_tiles) tile = n_tiles - 1;       // duplicate work, no deadlock
    const int m0 = tile * TILE_M;

    float* Sl = lds + wave * (2 * TILE_M * LDS_STRIDE);
    float* Hl = Sl + TILE_M * LDS_STRIDE;

    // ---- stage S tile (16 x 128) into LDS ----
#if USE_ASYNC_LDS
    // gfx1250 async copy: ASYNCcnt-tracked, bypasses VGPRs.
    for (int q = lane; q < TILE_M * (NF / 16); q += 32) {   // 16 rows x 8 b128
        const int r   = q >> 3;
        const int c16 = q & 7;
        __builtin_amdgcn_global_load_async_to_lds_b128(
            (g4i_t)(S + (size_t)(m0 + r) * NF + c16 * 16),
            (l4i_t)(Sl + r * LDS_STRIDE + c16 * 16),
            0, 0);
    }
    __builtin_amdgcn_s_wait_asynccnt(0);
#else
    for (int q = lane; q < TILE_M * (NF / 4); q += 32) {
        const int r  = q >> 5;         // row 0..15 (32 float4 per row)
        const int c4 = q & 31;
        const float4 v = *(const float4*)(S + (size_t)(m0 + r) * NF + c4 * 4);
        float* dst = Sl + r * LDS_STRIDE + c4 * 4;
        dst[0] = v.x; dst[1] = v.y; dst[2] = v.z; dst[3] = v.w;
    }
#endif
    __syncthreads();

    // ---- stage 1: H = silu(S @ W1 + b1), 8 n-tiles of 16 ----
    for (int nt = 0; nt < NF / 16; ++nt) {
        const int ncol = nt * 16 + l15;
        const float bias = b1[ncol];
        v8f acc;
        #pragma unroll
        for (int v = 0; v < 8; ++v) acc[v] = bias;

        #pragma unroll 4
        for (int k0 = 0; k0 < NF; k0 += 4) {
            v2f a, b;
            const int ka = k0 + 2 * half;
            a.x = Sl[l15 * LDS_STRIDE + ka];
            a.y = Sl[l15 * LDS_STRIDE + ka + 1];
            b.x = W1[(size_t)ka * NF + ncol];
            b.y = W1[(size_t)(ka + 1) * NF + ncol];
            acc = __builtin_amdgcn_wmma_f32_16x16x4_f32(
                false, a, false, b, (short)0, acc, false, false);
        }
        #pragma unroll
        for (int v = 0; v < 8; ++v)
            Hl[(v + 8 * half) * LDS_STRIDE + ncol] = fast_silu(acc[v]);
    }
    __syncthreads();

    // ---- stage 2: phi = H @ W2 + b2, 24 n-tiles of 16 ----
    for (int nt = 0; nt < N3F / 16; ++nt) {
        const int ncol = nt * 16 + l15;
        __builtin_prefetch(&W2[(size_t)(2 * half) * N3F + ncol + 16], 0, 1);
        const float bias = b2[ncol];
        v8f acc;
        #pragma unroll
        for (int v = 0; v < 8; ++v) acc[v] = bias;

        #pragma unroll 4
        for (int k0 = 0; k0 < NF; k0 += 4) {
            v2f a, b;
            const int ka = k0 + 2 * half;
            a.x = Hl[l15 * LDS_STRIDE + ka];
            a.y = Hl[l15 * LDS_STRIDE + ka + 1];
            b.x = W2[(size_t)ka * N3F + ncol];
            b.y = W2[(size_t)(ka + 1) * N3F + ncol];
            acc = __builtin_amdgcn_wmma_f32_16x16x4_f32(
                false, a, false, b, (short)0, acc, false, false);
        }
        #pragma unroll
        for (int v = 0; v < 8; ++v) {
            const int m = v + 8 * half;                     // C/D row layout
            phi[(size_t)(m0 + m) * N3F + ncol] = acc[v];
        }
    }
}

// ---------------------------------------------------------------------------
// Kernel 2: per-edge RBF linear + gather + elementwise + atomic scatter-add.
// One 128-thread block per edge; thread f handles feature channel f.
// phi_W split: [0:128)=vv, [128:256)=ss, [256:384)=vs.
// ---------------------------------------------------------------------------
__global__ __launch_bounds__(NF) void painn_edge(
    const int* __restrict__ idx_i, const int* __restrict__ idx_j,
    const float* __restrict__ rel_dir, const float* __restrict__ cut,
    const float* __restrict__ rbf, const float* __restrict__ Wr,
    const float* __restrict__ br, const float* __restrict__ phi,
    const float* __restrict__ vfeat,
    float* __restrict__ out_s, float* __restrict__ out_v)
{
    const int e = blockIdx.x;
    const int f = threadIdx.x;

    __shared__ float srbf[NR];
    if (f < NR) srbf[f] = rbf[(size_t)e * NR + f];
    __syncthreads();

    const int i = idx_i[e];
    const int j = idx_j[e];
    const float c  = cut[e];
    const float dx = rel_dir[3 * (size_t)e + 0];
    const float dy = rel_dir[3 * (size_t)e + 1];
    const float dz = rel_dir[3 * (size_t)e + 2];

    float wvv = br[f], wss = br[NF + f], wvs = br[2 * NF + f];
    #pragma unroll
    for (int r = 0; r < NR; ++r) {
        const float rv = srbf[r];
        wvv = fmaf(rv, Wr[(size_t)r * N3F + f],          wvv);
        wss = fmaf(rv, Wr[(size_t)r * N3F + NF + f],     wss);
        wvs = fmaf(rv, Wr[(size_t)r * N3F + 2 * NF + f], wvs);
    }

    const size_t pj  = (size_t)j * N3F;
    const float pvv = phi[pj + f]          * (wvv * c);
    const float pss = phi[pj + NF + f]     * (wss * c);
    const float pvs = phi[pj + 2 * NF + f] * (wvs * c);

    unsafeAtomicAdd(&out_s[(size_t)i * NF + f], pss);

    const size_t vj = ((size_t)j * NF + f) * 3;
    const size_t vi = ((size_t)i * NF + f) * 3;
    unsafeAtomicAdd(&out_v[vi + 0], fmaf(vfeat[vj + 0], pvv, pvs * dx));
    unsafeAtomicAdd(&out_v[vi + 1], fmaf(vfeat[vj + 1], pvv, pvs * dy));
    unsafeAtomicAdd(&out_v[vi + 2], fmaf(vfeat[vj + 2], pvv, pvs * dz));
}

// ---------------------------------------------------------------------------
extern "C" void kernel_launch(void* const* d_in, const int* in_sizes, int n_in,
                              void* d_out, int out_size, void* d_ws, size_t ws_size,
                              hipStream_t stream) {
    const int*   idx_i  = (const int*)  d_in[0];
    const int*   idx_j  = (const int*)  d_in[1];
    const float* reldir = (const float*)d_in[2];
    const float* cut    = (const float*)d_in[3];
    const float* rbf    = (const float*)d_in[4];
    const float* S      = (const float*)d_in[5];
    const float* vfeat  = (const float*)d_in[6];
    const float* W1     = (const float*)d_in[7];
    const float* b1     = (const float*)d_in[8];
    const float* W2     = (const float*)d_in[9];
    const float* b2     = (const float*)d_in[10];
    const float* Wr     = (const float*)d_in[11];
    const float* br     = (const float*)d_in[12];

    const int n_edges = in_sizes[0];
    const int n_nodes = in_sizes[5] / NF;

    float* phi   = (float*)d_ws;                         // n_nodes * 384 floats
    float* out_s = (float*)d_out;                        // [n_nodes, 128]
    float* out_v = out_s + (size_t)n_nodes * NF;         // [n_nodes, 128, 3]

    // residual init: out = input features (then scatter-add on top)
    (void)hipMemcpyAsync(out_s, S,     (size_t)n_nodes * NF * sizeof(float),
                         hipMemcpyDeviceToDevice, stream);
    (void)hipMemcpyAsync(out_v, vfeat, (size_t)n_nodes * NF * 3 * sizeof(float),
                         hipMemcpyDeviceToDevice, stream);

    const int n_tiles = n_nodes / TILE_M;                // 625
    const int blocks  = (n_tiles + WPB - 1) / WPB;
    painn_phi_wmma<<<blocks, 32 * WPB, 0, stream>>>(S, W1, b1, W2, b2, phi, n_nodes);

    painn_edge<<<n_edges, NF, 0, stream>>>(idx_i, idx_j, reldir, cut, rbf,
                                           Wr, br, phi, vfeat, out_s, out_v);
}